// Decoder_60902636257603
// MI455X (gfx1250) — compile-verified
//
#include <hip/hip_runtime.h>
#include <math.h>

#define HID 256
#define XS_STRIDE 260   // 256 + 4 pad: LDS column reads hit distinct banks
#define M_BLK 80        // rows per block: 80 | 100000, 5 subtiles of 16
#define GEMM1_LDS_BYTES (M_BLK * XS_STRIDE * 4)

typedef __attribute__((ext_vector_type(2))) float v2f;
typedef __attribute__((ext_vector_type(8))) float v8f;
typedef __attribute__((ext_vector_type(4))) unsigned int v4u;
typedef __attribute__((ext_vector_type(8))) int v8i;
typedef __attribute__((ext_vector_type(4))) int v4i;

// Use the Tensor Data Mover for the A-tile stage when the builtin exists
// (clang-23 / therock 6-arg form); otherwise fall back to cooperative staging.
#if defined(__gfx1250__) && defined(__has_builtin)
#  if __has_builtin(__builtin_amdgcn_tensor_load_to_lds) && \
      __has_builtin(__builtin_amdgcn_s_wait_tensorcnt)
#    define USE_TDM 1
#  endif
#endif
#ifndef USE_TDM
#  define USE_TDM 0
#endif

// ---------------- degree / norm ----------------
__global__ void k_init_deg(float* __restrict__ deg, int n) {
    int i = blockIdx.x * blockDim.x + threadIdx.x;
    if (i < n) deg[i] = 1.0f;  // self-loop
}

__global__ void k_count_deg(const int* __restrict__ dst, float* __restrict__ deg, int e) {
    int i = blockIdx.x * blockDim.x + threadIdx.x;
    if (i < e) unsafeAtomicAdd(&deg[dst[i]], 1.0f);
}

__global__ void k_finish_deg(float* __restrict__ dis, int n) {
    int i = blockIdx.x * blockDim.x + threadIdx.x;
    if (i < n) dis[i] = rsqrtf(dis[i]);   // deg >= 1 always
}

// ---------------- GEMM1: t1 = x @ W1 (fp32 WMMA 16x16x4) ----------------
// Block: 512 threads (16 waves), M-block = 80 rows staged in LDS (via TDM).
// Wave w owns output columns [w*16, w*16+16) and 5 M-subtiles -> 5 C frags.
// Each B fragment load (2 scalars) feeds 5 WMMAs -> W1 L2 traffic /5.
__global__ void __launch_bounds__(512)
k_gemm1(const float* __restrict__ x, const float* __restrict__ W1,
        float* __restrict__ t1) {
    extern __shared__ float xs[];          // M_BLK x XS_STRIDE, dynamic -> LDS offset 0

    const int tile = blockIdx.x;           // M block index
    const int t    = threadIdx.x;          // 0..511
    const int wave = t >> 5;               // 0..15
    const int lane = t & 31;
    const int half = lane >> 4;            // 0: K+0/K+1 lanes, 1: K+2/K+3 lanes
    const int l16  = lane & 15;

    const float* xrow = x + (size_t)tile * M_BLK * HID;

#if USE_TDM
    // ---- Tensor Data Mover: 2D tile (M_BLK x 256 f32), LDS pad 4 DW / 256 DW ----
    if (wave == 0) {
        unsigned long long ga = (unsigned long long)(const void*)xrow;
        v4u g0;
        g0.x = 0x1u;                                   // count=1 (valid descriptor)
        g0.y = 0u;                                     // lds_addr = 0 (dynamic LDS base)
        g0.z = (unsigned)(ga & 0xffffffffull);         // global_addr[31:0]
        g0.w = (unsigned)((ga >> 32) & 0x1ffffffull)   // global_addr[56:32]
             | (2u << 30);                             // type = 2 ("image")
        v8i g1;
        g1[0] = (int)((2u << 16)       // data_size = 2 -> 4 bytes
                    | (1u << 20)       // pad_enable
                    | (7u << 22)       // pad_interval code 7 -> every 256 DWORDs
                    | (3u << 25));     // pad_amount  code 3 -> 4 DWORDs (stride 260)
        g1[1] = (int)(256u << 16);                 // tensor_dim0 = 256 (bits 79:48)
        g1[2] = (int)((unsigned)M_BLK << 16);      // tensor_dim1 = M_BLK (bits 111:80)
        g1[3] = (int)(256u << 16);                 // tile_dim0 = 256 (bits 127:112)
        g1[4] = M_BLK;                             // tile_dim1 = M_BLK, tile_dim2 = 0
        g1[5] = 256;                               // tensor_dim0_stride = 256
        g1[6] = 0;                                 // stride hi / dim1_stride lo
        g1[7] = 0;
        v4i g2 = {0, 0, 0, 0};                     // group2: unused (2D tensor)
        v4i g3 = {0, 0, 0, 0};                     // group3: unused (2D tensor)
        v8i g4 = {0, 0, 0, 0, 0, 0, 0, 0};         // extra group (clang-23 form)
        __builtin_amdgcn_tensor_load_to_lds(g0, g1, g2, g3, g4, 0);
        __builtin_amdgcn_s_wait_tensorcnt(0);
    }
    __syncthreads();
#else
    // ---- fallback: cooperative, coalesced stage of the A tile ----
    for (int i = t; i < M_BLK * HID; i += 512) {
        int r = i >> 8, c = i & 255;
        xs[r * XS_STRIDE + c] = xrow[i];
    }
    __syncthreads();
#endif

    v8f acc[5];
    #pragma unroll
    for (int m = 0; m < 5; m++) acc[m] = (v8f){};

    const int col0 = wave * 16 + l16;
    for (int k = 0; k < HID; k += 4) {
        // B fragment (4x16): rows K = k + 2*half + {0,1}, col = col0
        const float* wp = W1 + (size_t)(k + 2 * half) * HID + col0;
        v2f b; b.x = wp[0]; b.y = wp[HID];

        #pragma unroll
        for (int m = 0; m < 5; m++) {
            // A fragment (16x4): row = m*16 + l16, K = k + 2*half + {0,1}
            const float* ap = &xs[(m * 16 + l16) * XS_STRIDE + k + 2 * half];
            v2f a; a.x = ap[0]; a.y = ap[1];
            acc[m] = __builtin_amdgcn_wmma_f32_16x16x4_f32(false, a, false, b,
                                                           (short)0, acc[m],
                                                           false, false);
        }
    }

    // C/D layout: VGPR v -> row v (lanes 0-15) / row v+8 (lanes 16-31), col = l16
    #pragma unroll
    for (int m = 0; m < 5; m++) {
        float* out = t1 + ((size_t)tile * M_BLK + m * 16 + half * 8) * HID + col0;
        #pragma unroll
        for (int v = 0; v < 8; v++) out[(size_t)v * HID] = acc[m][v];
    }
}

// ---------------- aggregation layer 1 ----------------
// agg1[i][c] = t1[i][c] * dis[i]^2   (self-loop term; also zero-inits agg1)
__global__ void k_selfinit_h(const float* __restrict__ t1, const float* __restrict__ dis,
                             float* __restrict__ agg1) {
    int i = blockIdx.x;            // node
    int c = threadIdx.x;           // channel
    float d = dis[i];
    agg1[(size_t)i * HID + c] = t1[(size_t)i * HID + c] * d * d;
}

// One wave per edge: 32 lanes x 8 channels, coalesced reads, hw f32 atomics.
// t1 (102 MB) is L2-resident (192 MB L2), so the random gather stays on-chip.
__global__ void k_scatter_h(const int* __restrict__ src, const int* __restrict__ dst,
                            const float* __restrict__ dis, const float* __restrict__ t1,
                            float* __restrict__ agg1, int e) {
    int wave = threadIdx.x >> 5;
    int lane = threadIdx.x & 31;
    int edge = blockIdx.x * 8 + wave;
    if (edge >= e) return;
    int s = src[edge], d = dst[edge];
    float nrm = dis[s] * dis[d];
    const float* in  = t1   + (size_t)s * HID;
    float*       out = agg1 + (size_t)d * HID;
    #pragma unroll
    for (int j = 0; j < 8; j++) {
        int c = lane + j * 32;
        unsafeAtomicAdd(&out[c], in[c] * nrm);
    }
}

// h1 = relu(agg1 + b1), in place
__global__ void k_bias_relu(float* __restrict__ agg1, const float* __restrict__ b1) {
    int i = blockIdx.x;
    int c = threadIdx.x;
    float v = agg1[(size_t)i * HID + c] + b1[c];
    agg1[(size_t)i * HID + c] = v > 0.0f ? v : 0.0f;
}

// ---------------- GEMM2: t2 = h1 @ W2 (N=3, wave-per-row) ----------------
__global__ void __launch_bounds__(256)
k_gemm2(const float* __restrict__ h1, const float* __restrict__ W2,
        float* __restrict__ t2) {
    int wave = threadIdx.x >> 5;
    int lane = threadIdx.x & 31;
    int row  = blockIdx.x * 8 + wave;
    const float* hr = h1 + (size_t)row * HID;
    float s0 = 0.f, s1 = 0.f, s2 = 0.f;
    for (int k = lane; k < HID; k += 32) {
        float hv = hr[k];
        s0 += hv * W2[k * 3 + 0];
        s1 += hv * W2[k * 3 + 1];
        s2 += hv * W2[k * 3 + 2];
    }
    #pragma unroll
    for (int off = 16; off > 0; off >>= 1) {
        s0 += __shfl_xor(s0, off, 32);
        s1 += __shfl_xor(s1, off, 32);
        s2 += __shfl_xor(s2, off, 32);
    }
    if (lane == 0) {
        t2[(size_t)row * 3 + 0] = s0;
        t2[(size_t)row * 3 + 1] = s1;
        t2[(size_t)row * 3 + 2] = s2;
    }
}

// ---------------- aggregation layer 2 (3 channels) ----------------
__global__ void k_selfinit_o(const float* __restrict__ t2, const float* __restrict__ dis,
                             float* __restrict__ agg2, int n3) {
    int idx = blockIdx.x * blockDim.x + threadIdx.x;
    if (idx < n3) {
        float d = dis[idx / 3];
        agg2[idx] = t2[idx] * d * d;
    }
}

__global__ void k_scatter_o(const int* __restrict__ src, const int* __restrict__ dst,
                            const float* __restrict__ dis, const float* __restrict__ t2,
                            float* __restrict__ agg2, int e) {
    int i = blockIdx.x * blockDim.x + threadIdx.x;
    if (i >= e) return;
    int s = src[i], d = dst[i];
    float nrm = dis[s] * dis[d];
    unsafeAtomicAdd(&agg2[(size_t)d * 3 + 0], t2[(size_t)s * 3 + 0] * nrm);
    unsafeAtomicAdd(&agg2[(size_t)d * 3 + 1], t2[(size_t)s * 3 + 1] * nrm);
    unsafeAtomicAdd(&agg2[(size_t)d * 3 + 2], t2[(size_t)s * 3 + 2] * nrm);
}

// ---------------- final: out = [agg2 + b2, init_state] @ fcW + fcb ----------------
__global__ void k_final(const float* __restrict__ agg2, const float* __restrict__ b2,
                        const float* __restrict__ init_state,
                        const float* __restrict__ fcW, const float* __restrict__ fcb,
                        float* __restrict__ out, int n) {
    int i = blockIdx.x * blockDim.x + threadIdx.x;
    if (i >= n) return;
    float h[11];
    #pragma unroll
    for (int j = 0; j < 3; j++) h[j] = agg2[(size_t)i * 3 + j] + b2[j];
    #pragma unroll
    for (int j = 0; j < 8; j++) h[3 + j] = init_state[(size_t)i * 8 + j];
    #pragma unroll
    for (int o = 0; o < 3; o++) {
        float acc = fcb[o];
        #pragma unroll
        for (int j = 0; j < 11; j++) acc += h[j] * fcW[j * 3 + o];
        out[(size_t)i * 3 + o] = acc;
    }
}

extern "C" void kernel_launch(void* const* d_in, const int* in_sizes, int n_in,
                              void* d_out, int out_size, void* d_ws, size_t ws_size,
                              hipStream_t stream) {
    const float* x          = (const float*)d_in[0];
    const int*   ei         = (const int*)  d_in[1];
    /* d_in[2] edge_attr unused by the reference */
    const float* init_state = (const float*)d_in[3];
    const float* W1         = (const float*)d_in[4];
    const float* b1         = (const float*)d_in[5];
    const float* W2         = (const float*)d_in[6];
    const float* b2         = (const float*)d_in[7];
    const float* fcW        = (const float*)d_in[8];
    const float* fcb        = (const float*)d_in[9];
    float*       out        = (float*)d_out;

    const int N = in_sizes[0] / HID;    // 100000
    const int E = in_sizes[1] / 2;      // 1600000
    const int* src = ei;
    const int* dst = ei + E;

    float* ws   = (float*)d_ws;
    float* dis  = ws;
    float* t1   = dis  + (size_t)N;
    float* agg1 = t1   + (size_t)N * HID;
    float* t2   = agg1 + (size_t)N * HID;
    float* agg2 = t2   + (size_t)N * 3;

    // degrees -> dis = rsqrt(deg) with self-loops
    k_init_deg  <<<(N + 255) / 256, 256, 0, stream>>>(dis, N);
    k_count_deg <<<(E + 255) / 256, 256, 0, stream>>>(dst, dis, E);
    k_finish_deg<<<(N + 255) / 256, 256, 0, stream>>>(dis, N);

    // layer 1
    k_gemm1     <<<N / M_BLK, 512, GEMM1_LDS_BYTES, stream>>>(x, W1, t1);
    k_selfinit_h<<<N, 256, 0, stream>>>(t1, dis, agg1);
    k_scatter_h <<<(E + 7) / 8, 256, 0, stream>>>(src, dst, dis, t1, agg1, E);
    k_bias_relu <<<N, 256, 0, stream>>>(agg1, b1);

    // layer 2
    k_gemm2     <<<N / 8, 256, 0, stream>>>(agg1, W2, t2);
    k_selfinit_o<<<(N * 3 + 255) / 256, 256, 0, stream>>>(t2, dis, agg2, N * 3);
    k_scatter_o <<<(E + 255) / 256, 256, 0, stream>>>(src, dst, dis, t2, agg2, E);

    // decoder head
    k_final     <<<(N + 255) / 256, 256, 0, stream>>>(agg2, b2, init_state, fcW, fcb, out, N);
}